// AttentionMechanisms_61151744360763
// MI455X (gfx1250) — compile-verified
//
#include <hip/hip_runtime.h>

typedef __attribute__((ext_vector_type(2))) float v2f;
typedef __attribute__((ext_vector_type(8))) float v8f;

#define DIM 512
#define TQ 32
#define TK 32
#define DC 128
#define LSTRIDE 132   // 128 + 4 pad: rows 16B-aligned, banks rotate by 4/row
#define NCHUNK (DIM / DC)

__device__ __forceinline__ float fast_tanh(float x) {
#if __has_builtin(__builtin_amdgcn_tanhf)
  return __builtin_amdgcn_tanhf(x);       // gfx1250 v_tanh_f32
#else
  return tanhf(x);
#endif
}

// Async copy 16 bytes global -> LDS (per lane), tracked by ASYNCcnt.
__device__ __forceinline__ void async_copy_b128(unsigned lds_byte, const float* g) {
  asm volatile("global_load_async_to_lds_b128 %0, %1, off"
               :: "v"(lds_byte), "v"(g)
               : "memory");
}

template <int N>
__device__ __forceinline__ void wait_async() {
#if __has_builtin(__builtin_amdgcn_s_wait_asynccnt)
  __builtin_amdgcn_s_wait_asynccnt(N);
#else
  asm volatile("s_wait_asynccnt %0" :: "i"(N) : "memory");
#endif
}

// out[r][n] = sum_d X[r][d] * W[n][d] + bias[n]
// One wave (32 lanes) computes one 16x16 tile via V_WMMA_F32_16X16X4_F32.
__global__ __launch_bounds__(32)
void proj_wmma_kernel(const float* __restrict__ X, const float* __restrict__ W,
                      const float* __restrict__ bias, float* __restrict__ out) {
  const int lane = threadIdx.x;            // 0..31
  const int n0   = blockIdx.x * 16;        // output col tile
  const int m0   = blockIdx.y * 16;        // output row tile
  const int half = lane >> 4;              // 0 or 1 (lane half)
  const int l16  = lane & 15;
  const int klo  = half * 2;               // A/B: lane-half picks K pair {0,1} vs {2,3}

  const float* Arow = X + (size_t)(m0 + l16) * DIM;   // A[m][k] = X[m][k]
  const float* Brow = W + (size_t)(n0 + l16) * DIM;   // B[k][n] = W[n][k]

  v8f c;
  const float bz = bias[n0 + l16];         // C/D: N = lane%16 for every VGPR
  #pragma unroll
  for (int j = 0; j < 8; ++j) c[j] = bz;

  #pragma unroll 4
  for (int k = 0; k < DIM; k += 4) {
    v2f a, b;
    a.x = Arow[k + klo];
    a.y = Arow[k + klo + 1];
    b.x = Brow[k + klo];
    b.y = Brow[k + klo + 1];
    c = __builtin_amdgcn_wmma_f32_16x16x4_f32(false, a, false, b,
                                              (short)0, c, false, false);
  }

  // D layout: VGPR j -> row m0 + j + half*8, col n0 + lane%16
  float* orow = out + (size_t)(m0 + half * 8) * DIM + (n0 + l16);
  #pragma unroll
  for (int j = 0; j < 8; ++j) orow[(size_t)j * DIM] = c[j];
}

// Issue one chunk's async global->LDS copies: 32x128 floats for q and k each.
// 8 async B128 ops per thread per chunk.
__device__ __forceinline__ void issue_chunk(const float* qbase, const float* kbase,
                                            int d0, float* bq, float* bk, int t) {
  #pragma unroll
  for (int i = 0; i < 4; ++i) {
    int id = t + i * 256;          // 0..1023
    int r  = id >> 5;              // 0..31  (row)
    int c  = (id & 31) << 2;       // 0..124 (col, multiple of 4 -> 16B aligned)
    async_copy_b128((unsigned)(uintptr_t)&bq[r * LSTRIDE + c],
                    qbase + (size_t)r * DIM + d0 + c);
    async_copy_b128((unsigned)(uintptr_t)&bk[r * LSTRIDE + c],
                    kbase + (size_t)r * DIM + d0 + c);
  }
}

// scores[b,q,k] = sum_d v[d] * tanh(qt[b,q,d] + kt[b,k,d])
// 32x32 tile per block, 2q x 2k register tile per thread, double-buffered
// async LDS chunks over d.
__global__ __launch_bounds__(256)
void scores_kernel(const float* __restrict__ qt, const float* __restrict__ kt,
                   const float* __restrict__ vv, float* __restrict__ out) {
  __shared__ float bq[2][TQ * LSTRIDE];
  __shared__ float bk[2][TK * LSTRIDE];
  __shared__ float sv[DIM];

  const int t  = threadIdx.x;
  const int b  = blockIdx.z;
  const int q0 = blockIdx.y * TQ;
  const int k0 = blockIdx.x * TK;

  const float* qbase = qt + ((size_t)b * 128 + q0) * DIM;
  const float* kbase = kt + ((size_t)b * 256 + k0) * DIM;

  // preload all of v once
  sv[t]       = vv[t];
  sv[t + 256] = vv[t + 256];

  const int tq = t >> 4;       // 0..15: q rows {tq, tq+16}
  const int tk = t & 15;       // 0..15: k cols {tk, tk+16}

  float acc00 = 0.f, acc01 = 0.f, acc10 = 0.f, acc11 = 0.f;

  issue_chunk(qbase, kbase, 0, bq[0], bk[0], t);

  for (int c = 0; c < NCHUNK; ++c) {
    const int cur = c & 1;
    if (c + 1 < NCHUNK) {
      issue_chunk(qbase, kbase, (c + 1) * DC, bq[cur ^ 1], bk[cur ^ 1], t);
      wait_async<8>();   // 8 outstanding = next chunk; current chunk complete
    } else {
      wait_async<0>();
    }
    __syncthreads();     // all waves' async data visible (also covers sv preload)

    const float* qr0 = &bq[cur][tq * LSTRIDE];
    const float* qr1 = &bq[cur][(tq + 16) * LSTRIDE];
    const float* kr0 = &bk[cur][tk * LSTRIDE];
    const float* kr1 = &bk[cur][(tk + 16) * LSTRIDE];
    const float* vp  = &sv[c * DC];

    #pragma unroll 4
    for (int d = 0; d < DC; ++d) {
      float vd = vp[d];
      float qa = qr0[d];
      float qb = qr1[d];
      float ka = kr0[d];
      float kb = kr1[d];
      acc00 += vd * fast_tanh(qa + ka);
      acc01 += vd * fast_tanh(qa + kb);
      acc10 += vd * fast_tanh(qb + ka);
      acc11 += vd * fast_tanh(qb + kb);
    }
    __syncthreads();     // done reading before buffer is overwritten next iter
  }

  float* o0 = out + ((size_t)b * 128 + q0 + tq) * 256 + k0;
  o0[tk]      = acc00;
  o0[tk + 16] = acc01;
  float* o1 = out + ((size_t)b * 128 + q0 + tq + 16) * 256 + k0;
  o1[tk]      = acc10;
  o1[tk + 16] = acc11;
}

extern "C" void kernel_launch(void* const* d_in, const int* in_sizes, int n_in,
                              void* d_out, int out_size, void* d_ws, size_t ws_size,
                              hipStream_t stream) {
  (void)in_sizes; (void)n_in; (void)out_size; (void)ws_size;
  const float* query = (const float*)d_in[0];  // [8,128,512]
  const float* key   = (const float*)d_in[1];  // [8,256,512]
  const float* W1    = (const float*)d_in[2];  // [512,512]
  const float* b1    = (const float*)d_in[3];  // [512]
  const float* W2    = (const float*)d_in[4];  // [512,512]
  const float* b2    = (const float*)d_in[5];  // [512]
  const float* v     = (const float*)d_in[6];  // [512]
  float* scores = (float*)d_out;               // [8,128,256]

  float* qt = (float*)d_ws;                    // 1024*512 f32 = 2 MB
  float* kt = qt + (size_t)1024 * DIM;         // 2048*512 f32 = 4 MB

  // qt = query @ W2^T + b2   (rows = 8*128 = 1024)
  proj_wmma_kernel<<<dim3(DIM / 16, 1024 / 16), dim3(32), 0, stream>>>(query, W2, b2, qt);
  // kt = key @ W1^T + b1     (rows = 8*256 = 2048)
  proj_wmma_kernel<<<dim3(DIM / 16, 2048 / 16), dim3(32), 0, stream>>>(key, W1, b1, kt);

  // scores = sum_d v[d] * tanh(qt + kt)
  scores_kernel<<<dim3(256 / TK, 128 / TQ, 8), dim3(256), 0, stream>>>(qt, kt, v, scores);
}